// SpatialTransformer2d_51926154609284
// MI455X (gfx1250) — compile-verified
//
#include <hip/hip_runtime.h>
#include <hip/hip_bf16.h>
#include <math.h>

typedef float v2f __attribute__((ext_vector_type(2)));
typedef float v8f __attribute__((ext_vector_type(8)));

__device__ __forceinline__ int clampi(int v, int lo, int hi) {
    return v < lo ? lo : (v > hi ? hi : v);
}

// ---------------------------------------------------------------- conv1 (1->16, 3x3, s1, p1), direct, branchless
__global__ __launch_bounds__(256) void conv1_kernel(const float* __restrict__ in,
                                                    const float* __restrict__ w,
                                                    float* __restrict__ out) {
    __shared__ float wsm[144];
    int tid = threadIdx.x;
    if (tid < 144) wsm[tid] = w[tid];
    __syncthreads();
    int idx = blockIdx.x * 256 + tid;          // 768*768 pixels exactly
    int oy = idx / 768, ox = idx % 768;
    float v[9];
#pragma unroll
    for (int ky = 0; ky < 3; ++ky)
#pragma unroll
        for (int kx = 0; kx < 3; ++kx) {
            int iy = oy + ky - 1, ix = ox + kx - 1;
            bool ok = ((unsigned)iy < 768u) & ((unsigned)ix < 768u);
            float t = in[clampi(iy, 0, 767) * 768 + clampi(ix, 0, 767)]; // always in-bounds
            v[ky * 3 + kx] = ok ? t : 0.f;
        }
#pragma unroll
    for (int oc = 0; oc < 16; ++oc) {
        float s = 0.f;
#pragma unroll
        for (int t = 0; t < 9; ++t) s = fmaf(wsm[oc * 9 + t], v[t], s);
        out[(size_t)oc * 589824 + idx] = s;
    }
}

// ---------------------------------------------------------------- BN stats, two-stage deterministic tree
#define SPB 64   // partial blocks per channel

__global__ __launch_bounds__(256) void statsA_kernel(const float* __restrict__ x, int hw,
                                                     float* __restrict__ partial) {
    int c = blockIdx.y, b = blockIdx.x;
    const float* p = x + (size_t)c * hw;
    float s = 0.f, ss = 0.f;
    for (int i = b * 256 + threadIdx.x; i < hw; i += SPB * 256) {
        float v = p[i]; s += v; ss = fmaf(v, v, ss);
    }
    __shared__ float sm[256], sm2[256];
    sm[threadIdx.x] = s; sm2[threadIdx.x] = ss;
    __syncthreads();
    for (int off = 128; off > 0; off >>= 1) {
        if (threadIdx.x < off) { sm[threadIdx.x] += sm[threadIdx.x + off]; sm2[threadIdx.x] += sm2[threadIdx.x + off]; }
        __syncthreads();
    }
    if (threadIdx.x == 0) {
        partial[(c * SPB + b) * 2]     = sm[0];
        partial[(c * SPB + b) * 2 + 1] = sm2[0];
    }
}

__global__ __launch_bounds__(SPB) void statsB_kernel(const float* __restrict__ partial, int hw,
                                                     float* __restrict__ mean,
                                                     float* __restrict__ istd) {
    int c = blockIdx.x;
    __shared__ float sm[SPB], sm2[SPB];
    sm[threadIdx.x]  = partial[(c * SPB + threadIdx.x) * 2];
    sm2[threadIdx.x] = partial[(c * SPB + threadIdx.x) * 2 + 1];
    __syncthreads();
    for (int off = SPB / 2; off > 0; off >>= 1) {
        if (threadIdx.x < off) { sm[threadIdx.x] += sm[threadIdx.x + off]; sm2[threadIdx.x] += sm2[threadIdx.x + off]; }
        __syncthreads();
    }
    if (threadIdx.x == 0) {
        float m = sm[0] / (float)hw;
        float var = sm2[0] / (float)hw - m * m;
        mean[c] = m;
        istd[c] = rsqrtf(var + 1e-5f);
    }
}

__global__ __launch_bounds__(256) void bnrelu_kernel(float* __restrict__ x, int hw,
                                                     const float* __restrict__ mean,
                                                     const float* __restrict__ istd) {
    int c = blockIdx.y;
    int i = c * hw + blockIdx.x * 256 + threadIdx.x;
    float v = (x[i] - mean[c]) * istd[c];
    x[i] = v > 0.f ? v : 0.f;
}

// ---------------------------------------------------------------- WMMA implicit-GEMM conv (3x3, stride2, pad1)
// Branchless im2col fetch: clamped address, unconditional load, cndmask to zero.
template <int WIN>
__device__ __forceinline__ float im2col_fetch(const float* __restrict__ x, int k, int oy, int ox) {
    int c = k / 9; int r = k - 9 * c; int ky = r / 3; int kx = r - 3 * ky;
    int iy = oy * 2 + ky - 1, ix = ox * 2 + kx - 1;
    bool ok = ((unsigned)iy < (unsigned)WIN) & ((unsigned)ix < (unsigned)WIN);
    float t = x[((size_t)c * WIN + clampi(iy, 0, WIN - 1)) * WIN + clampi(ix, 0, WIN - 1)];
    return ok ? t : 0.f;
}

// D(M x 16) = W(M x K) * im2col(K x 16), M = 16*MTILES, K = 9*CIN, via V_WMMA_F32_16X16X4_F32.
// Weights staged in LDS transposed as [k][m]; chunked so static LDS <= ~36KB.
template <int CIN, int WIN, int WOUT, int MTILES>
__global__ __launch_bounds__(256) void conv_wmma_kernel(const float* __restrict__ x,
                                                        const float* __restrict__ w,
                                                        float* __restrict__ y) {
    constexpr int K = CIN * 9;
    constexpr int M = MTILES * 16;
    constexpr int KC = (K * M * 4 > 65536) ? (K / 2) : K;   // conv2: 144, conv3: 144 (2 chunks)
    constexpr int NOUT = WOUT * WOUT;
    __shared__ float wlds[KC * M];

    int tid = threadIdx.x;
    int lane = tid & 31;
    int tile = blockIdx.x * 8 + (tid >> 5);           // grid sized exactly: full-EXEC waves only
    int px = tile * 16 + (lane & 15);                 // output pixel (N column)
    int oy = px / WOUT, ox = px % WOUT;
    int koff = (lane >> 4) * 2;                       // lanes 16-31 carry K+2, K+3
    int am = lane & 15;                               // A row within a 16-row tile
    v8f acc[MTILES] = {};

    for (int kc = 0; kc < K; kc += KC) {
        __syncthreads();
        for (int i = tid; i < KC * M; i += 256) {
            int kl = i / M, m = i - kl * M;
            wlds[i] = w[(size_t)m * K + kc + kl];     // wlds[kl*M + m]
        }
        __syncthreads();
        for (int kb = 0; kb < KC; kb += 4) {
            int kg = kc + kb + koff;
            float b0 = im2col_fetch<WIN>(x, kg, oy, ox);
            float b1 = im2col_fetch<WIN>(x, kg + 1, oy, ox);
            v2f bv = {b0, b1};
            int kl = kb + koff;
#pragma unroll
            for (int t = 0; t < MTILES; ++t) {
                int m = t * 16 + am;
                v2f av = {wlds[kl * M + m], wlds[(kl + 1) * M + m]};
                acc[t] = __builtin_amdgcn_wmma_f32_16x16x4_f32(
                    false, av, false, bv, (short)0, acc[t], false, false);
            }
        }
    }
    int rh = (lane >> 4) * 8;                         // D: vgpr j holds rows rh+j
#pragma unroll
    for (int t = 0; t < MTILES; ++t)
#pragma unroll
        for (int j = 0; j < 8; ++j)
            y[(size_t)(t * 16 + rh + j) * NOUT + px] = acc[t][j];
}

// ---------------------------------------------------------------- five 2x2/s2 tanh heads
__global__ __launch_bounds__(256) void heads_kernel(const float* __restrict__ f3,
    const float* __restrict__ pw, const float* __restrict__ pb,
    const float* __restrict__ tw, const float* __restrict__ tb,
    const float* __restrict__ shw, const float* __restrict__ shb,
    const float* __restrict__ scw, const float* __restrict__ scb,
    const float* __restrict__ tiw, const float* __restrict__ tib,
    float* __restrict__ psi_m, float* __restrict__ theta_m,
    float* __restrict__ scale_m, float* __restrict__ tilt_m,
    float* __restrict__ shift_m) {
    int n = blockIdx.x * 256 + threadIdx.x;           // 9216 exactly
    int yy = n / 96, xx = n % 96;
    const float* base = f3 + (size_t)(yy * 2) * 192 + xx * 2;
    float sp = 0.f, st = 0.f, s0 = 0.f, s1 = 0.f, ssc = 0.f, sti = 0.f;
    for (int c = 0; c < 64; ++c) {
        const float* fp = base + (size_t)c * 36864;
        float f00 = fp[0], f01 = fp[1], f10 = fp[192], f11 = fp[193];
        int k = c * 4;
        sp  = fmaf(pw[k], f00, fmaf(pw[k+1], f01, fmaf(pw[k+2], f10, fmaf(pw[k+3], f11, sp))));
        st  = fmaf(tw[k], f00, fmaf(tw[k+1], f01, fmaf(tw[k+2], f10, fmaf(tw[k+3], f11, st))));
        s0  = fmaf(shw[k], f00, fmaf(shw[k+1], f01, fmaf(shw[k+2], f10, fmaf(shw[k+3], f11, s0))));
        s1  = fmaf(shw[256+k], f00, fmaf(shw[256+k+1], f01, fmaf(shw[256+k+2], f10, fmaf(shw[256+k+3], f11, s1))));
        ssc = fmaf(scw[k], f00, fmaf(scw[k+1], f01, fmaf(scw[k+2], f10, fmaf(scw[k+3], f11, ssc))));
        sti = fmaf(tiw[k], f00, fmaf(tiw[k+1], f01, fmaf(tiw[k+2], f10, fmaf(tiw[k+3], f11, sti))));
    }
    psi_m[n]   = 3.14159f * tanhf(sp + pb[0]);
    theta_m[n] = 3.14159f * tanhf(st + tb[0]);
    shift_m[n]        = 0.2f * tanhf(s0 + shb[0]);
    shift_m[9216 + n] = 0.2f * tanhf(s1 + shb[1]);
    float sc = 1.f + 0.25f * tanhf(ssc + scb[0]);
    scale_m[n] = fminf(fmaxf(sc, 0.8f), 1.25f);
    float ti = 1.f + 0.8f * tanhf(sti + tib[0]);
    tilt_m[n] = fminf(fmaxf(ti, 1.0f), 1.8f);
}

// ---------------------------------------------------------------- build 2x3 transforms + aff output
__global__ __launch_bounds__(256) void transform_kernel(const float* __restrict__ psi_m,
    const float* __restrict__ theta_m, const float* __restrict__ scale_m,
    const float* __restrict__ tilt_m, const float* __restrict__ shift_m,
    float* __restrict__ trans, float* __restrict__ aff_out) {
    int n = blockIdx.x * 256 + threadIdx.x;           // 9216 exactly
    float p = psi_m[n], th = theta_m[n], s = scale_m[n], t = tilt_m[n];
    float cp = cosf(p), sp = sinf(p);
    float ct = cosf(th), st = sinf(th);
    float it = 1.f / t;
    float m00 = t * ct,  m01 = -t * st;               // T @ Rt
    float m10 = it * st, m11 = it * ct;
    float a00 = s * (cp * m00 - sp * m10);            // S @ Rp @ (T @ Rt)
    float a01 = s * (cp * m01 - sp * m11);
    float a10 = s * (sp * m00 + cp * m10);
    float a11 = s * (sp * m01 + cp * m11);
    // reference's shift.reshape(-1,2,1): flat (C,HW) order read at 2n, 2n+1
    float b0 = shift_m[2 * n], b1 = shift_m[2 * n + 1];
    float* tr = trans + (size_t)n * 6;
    tr[0] = a00; tr[1] = a01; tr[2] = b0;
    tr[3] = a10; tr[4] = a11; tr[5] = b1;
    float cx = 4.f + 8.f * (float)(n % 96);
    float cy = 4.f + 8.f * (float)(n / 96);
    float* ao = aff_out + (size_t)n * 6;
    ao[0] = 32.f * a00; ao[1] = 32.f * a01; ao[2] = 32.f * b0 + cx;
    ao[3] = 32.f * a10; ao[4] = 32.f * a11; ao[5] = 32.f * b1 + cy;
}

// ---------------------------------------------------------------- bilinear patch sampler (branchless gathers)
__global__ __launch_bounds__(256) void sampler_kernel(const float* __restrict__ img,
                                                      const float* __restrict__ trans,
                                                      float* __restrict__ out) {
    int gid = blockIdx.x * 256 + threadIdx.x;         // 9216*1024 exactly
    int n = gid >> 10;
    int p = gid & 1023;
    int h = p >> 5, w = p & 31;
    const float* tr = trans + (size_t)n * 6;
    float X = ((float)w + 0.5f) * (2.f / 32.f) - 1.f;
    float Y = ((float)h + 0.5f) * (2.f / 32.f) - 1.f;
    float g0 = (tr[0] * X + tr[1] * Y + tr[2]) / 4.5f;
    float g1 = (tr[3] * X + tr[4] * Y + tr[5]) / 4.5f;
    float ixf = ((g0 + 1.f) * 144.f - 1.f) * 0.5f;
    float iyf = ((g1 + 1.f) * 144.f - 1.f) * 0.5f;
    float x0f = floorf(ixf), y0f = floorf(iyf);
    float wx = ixf - x0f, wy = iyf - y0f;
    int x0 = (int)x0f, y0 = (int)y0f;
    int by = (n / 96) * 8 - 68, bx = (n % 96) * 8 - 68;   // patch(r,c) -> padded img
    auto g = [&](int yy, int xx) -> float {
        bool ok = ((unsigned)yy < 144u) & ((unsigned)xx < 144u);
        int gy = by + clampi(yy, 0, 143), gx = bx + clampi(xx, 0, 143);
        ok = ok & ((unsigned)gy < 768u) & ((unsigned)gx < 768u);
        float t = img[clampi(gy, 0, 767) * 768 + clampi(gx, 0, 767)]; // always in-bounds
        return ok ? t : 0.f;
    };
    float v00 = g(y0, x0), v01 = g(y0, x0 + 1);
    float v10 = g(y0 + 1, x0), v11 = g(y0 + 1, x0 + 1);
    out[gid] = (1.f - wy) * ((1.f - wx) * v00 + wx * v01) +
               wy * ((1.f - wx) * v10 + wx * v11);
}

// ---------------------------------------------------------------- launcher
extern "C" void kernel_launch(void* const* d_in, const int* in_sizes, int n_in,
                              void* d_out, int out_size, void* d_ws, size_t ws_size,
                              hipStream_t stream) {
    (void)in_sizes; (void)n_in; (void)out_size; (void)ws_size;
    const float* img = (const float*)d_in[0];
    const float* w1  = (const float*)d_in[1];
    const float* w2  = (const float*)d_in[2];
    const float* w3  = (const float*)d_in[3];
    const float* pw  = (const float*)d_in[4];
    const float* pb  = (const float*)d_in[5];
    const float* tw  = (const float*)d_in[6];
    const float* tb  = (const float*)d_in[7];
    const float* shw = (const float*)d_in[8];
    const float* shb = (const float*)d_in[9];
    const float* scw = (const float*)d_in[10];
    const float* scb = (const float*)d_in[11];
    const float* tiw = (const float*)d_in[12];
    const float* tib = (const float*)d_in[13];
    float* out = (float*)d_out;
    float* ws  = (float*)d_ws;

    float* F1 = ws;                        // 16*768*768 = 9437184
    float* F2 = ws + 9437184;              // 32*384*384 = 4718592 (ends 14155776)
    float* F3 = ws;                        // 64*192*192 = 2359296, reuses F1 (dead by then)
    float* stats = ws + 14155776;          // 224 floats
    float* mean1 = stats,        *istd1 = stats + 16;
    float* mean2 = stats + 32,   *istd2 = stats + 64;
    float* mean3 = stats + 96,   *istd3 = stats + 160;
    float* maps = ws + 14156000;           // 6*9216
    float* psi_m = maps, *theta_m = maps + 9216, *scale_m = maps + 2 * 9216;
    float* tilt_m = maps + 3 * 9216, *shift_m = maps + 4 * 9216;
    float* trans = ws + 14156000 + 6 * 9216;     // 9216*6  (ends 14266592)
    float* partial = ws + 14266592;              // 64*SPB*2 = 8192 floats

    conv1_kernel<<<2304, 256, 0, stream>>>(img, w1, F1);
    statsA_kernel<<<dim3(SPB, 16), 256, 0, stream>>>(F1, 589824, partial);
    statsB_kernel<<<16, SPB, 0, stream>>>(partial, 589824, mean1, istd1);
    bnrelu_kernel<<<dim3(2304, 16), 256, 0, stream>>>(F1, 589824, mean1, istd1);

    conv_wmma_kernel<16, 768, 384, 2><<<1152, 256, 0, stream>>>(F1, w2, F2);
    statsA_kernel<<<dim3(SPB, 32), 256, 0, stream>>>(F2, 147456, partial);
    statsB_kernel<<<32, SPB, 0, stream>>>(partial, 147456, mean2, istd2);
    bnrelu_kernel<<<dim3(576, 32), 256, 0, stream>>>(F2, 147456, mean2, istd2);

    conv_wmma_kernel<32, 384, 192, 4><<<288, 256, 0, stream>>>(F2, w3, F3);
    statsA_kernel<<<dim3(SPB, 64), 256, 0, stream>>>(F3, 36864, partial);
    statsB_kernel<<<64, SPB, 0, stream>>>(partial, 36864, mean3, istd3);
    bnrelu_kernel<<<dim3(144, 64), 256, 0, stream>>>(F3, 36864, mean3, istd3);

    heads_kernel<<<36, 256, 0, stream>>>(F3, pw, pb, tw, tb, shw, shb, scw, scb, tiw, tib,
                                         psi_m, theta_m, scale_m, tilt_m, shift_m);
    transform_kernel<<<36, 256, 0, stream>>>(psi_m, theta_m, scale_m, tilt_m, shift_m,
                                             trans, out + 9437184);
    sampler_kernel<<<36864, 256, 0, stream>>>(img, trans, out);
}